// SOAPDescriptor_7275674599659
// MI455X (gfx1250) — compile-verified
//
#include <hip/hip_runtime.h>
#include <hip/hip_bf16.h>
#include <stdint.h>

// Problem constants (from the reference)
#define NATOMS 3072
#define NMAX   8
#define LMAXP1 7
#define ROWLEN (NMAX * NATOMS + LMAXP1)   // 24583 floats per output row
#define RCUT   6.0f
#define ALPHA  2.0f                        // 0.5 / sigma^2, sigma = 0.5
#define JTILE  256                         // j atoms per block (= blockDim.x)

// Low 32 bits of a flat LDS pointer are the LDS byte offset (aperture lives
// in bits [63:32] on CDNA5), so truncation yields the DS address for the
// async-to-LDS destination VGPR.
__device__ __forceinline__ uint32_t lds_byte_addr(const void* p) {
    return (uint32_t)(uintptr_t)p;
}

__global__ void __launch_bounds__(JTILE)
soap_radial_kernel(const float* __restrict__ positions,
                   const float* __restrict__ cell,
                   float* __restrict__ out) {
    __shared__ __align__(16) float shpos[JTILE * 3];   // AoS xyz tile, 3072 B

    const int t     = threadIdx.x;
    const int jbase = blockIdx.x * JTILE;
    const int i     = blockIdx.y;

    // ---- Async-stage this block's j-tile of positions into LDS ----------
    // 256 atoms * 12 B = 3072 B = 192 x b128 transfers (lanes 0..191).
    if (t < (JTILE * 3 * 4) / 16) {
        uint64_t gaddr = (uint64_t)(uintptr_t)(positions + (size_t)jbase * 3) +
                         (uint64_t)t * 16u;
        uint32_t laddr = lds_byte_addr(&shpos[0]) + (uint32_t)t * 16u;
        asm volatile("global_load_async_to_lds_b128 %0, %1, off"
                     :: "v"(laddr), "v"(gaddr) : "memory");
    }
    asm volatile("s_wait_asynccnt 0x0" ::: "memory");
    __syncthreads();   // make all waves' async data visible before ds_load

    // ---- Uniform (per-block) values: central atom i, box from cell diag --
    const float xi = positions[3 * i + 0];
    const float yi = positions[3 * i + 1];
    const float zi = positions[3 * i + 2];
    const float bx = cell[0], by = cell[4], bz = cell[8];    // diagonal cell
    const float ibx = 1.0f / bx, iby = 1.0f / by, ibz = 1.0f / bz;

    // ---- Per-pair computation -------------------------------------------
    const int j  = jbase + t;
    const float xj = shpos[3 * t + 0];
    const float yj = shpos[3 * t + 1];
    const float zj = shpos[3 * t + 2];

    // minimum-image convention (round-to-nearest-even matches jnp.round)
    float dx = xi - xj;  dx -= bx * rintf(dx * ibx);
    float dy = yi - yj;  dy -= by * rintf(dy * iby);
    float dz = zi - zj;  dz -= bz * rintf(dz * ibz);

    const float d2 = fmaf(dx, dx, fmaf(dy, dy, dz * dz)) + 1e-10f;
    const float d  = sqrtf(d2);
    const bool  m  = (d < RCUT) && (d > 0.1f);

    // g_n = (d/rcut)^n * exp(-alpha*d^2), zero outside the neighbor shell
    float g  = m ? __expf(-ALPHA * d2) : 0.0f;
    const float tr = d * (1.0f / RCUT);

    // 8 coalesced non-temporal stores per pair: out[i, n*N + j]
    float* orow = out + (size_t)i * ROWLEN + (size_t)j;
#pragma unroll
    for (int n = 0; n < NMAX; ++n) {
        __builtin_nontemporal_store(g, orow + (size_t)n * NATOMS);
        g *= tr;
    }
}

// Zero the 7 trailing "angular" columns of every row (harness poisons d_out).
__global__ void soap_angular_zero_kernel(float* __restrict__ out) {
    const int idx = blockIdx.x * blockDim.x + threadIdx.x;
    if (idx < NATOMS * LMAXP1) {
        const int i = idx / LMAXP1;
        const int l = idx % LMAXP1;
        __builtin_nontemporal_store(
            0.0f, out + (size_t)i * ROWLEN + (size_t)(NMAX * NATOMS) + l);
    }
}

extern "C" void kernel_launch(void* const* d_in, const int* in_sizes, int n_in,
                              void* d_out, int out_size, void* d_ws, size_t ws_size,
                              hipStream_t stream) {
    const float* positions = (const float*)d_in[0];   // [3072,3] f32
    // d_in[1] = atom_types (unused by the math)
    const float* cell      = (const float*)d_in[2];   // [3,3] f32 (diagonal)
    float*       out       = (float*)d_out;           // [3072, 24583] f32

    dim3 grid(NATOMS / JTILE, NATOMS, 1);             // (12 j-tiles, 3072 rows)
    soap_radial_kernel<<<grid, JTILE, 0, stream>>>(positions, cell, out);

    const int tail = NATOMS * LMAXP1;                 // 21504 zeros
    soap_angular_zero_kernel<<<(tail + 255) / 256, 256, 0, stream>>>(out);
}